// PatchRepulsionLoss_90855738179772
// MI455X (gfx1250) — compile-verified
//
#include <hip/hip_runtime.h>
#include <stdint.h>

#define NPTS         2048
#define NPATCH       32
#define PPP          64                  // points per patch
#define FLOATS_PER_B (NPTS * 3)          // 6144 floats per batch element
#define VEC4_PER_B   (FLOATS_PER_B / 4)  // 1536 float4 per batch element (24 KiB)
#define THREADS      256
#define V4_PER_THR   (VEC4_PER_B / THREADS)   // 6
#define V4_PER_PATCH 48                  // float4 per patch (768 B)
#define PATCH_STRIDE 49                  // float4 stride in LDS (784 B, bank skew)

__global__ __launch_bounds__(THREADS)
void patch_repulsion_kernel(const float* __restrict__ pcs,
                            float* __restrict__ partials,   // per-block unscaled sums
                            float* __restrict__ out,        // atomic fallback target
                            float scale) {
    // Padded staging buffer: 32 patches * 49 float4 = 25,088 B
    __shared__ __align__(16) float sdataf[NPATCH * PATCH_STRIDE * 4];
    __shared__ float smean[NPATCH][4];
    __shared__ float sstd[NPATCH][4];
    __shared__ float swsum[THREADS / 32];

    const int tid = threadIdx.x;
    const int b   = blockIdx.x;

    // ---------------- Phase 1: async global -> LDS staging (CDNA5 path) -----
    // Fully coalesced 128b streaming from HBM; per-lane LDS destinations are
    // skewed by 16B per patch so phase-2 reads hit distinct bank groups.
    const float4* gsrc = reinterpret_cast<const float4*>(pcs) + (size_t)b * VEC4_PER_B;
#pragma unroll
    for (int k = 0; k < V4_PER_THR; ++k) {
        const int idx   = tid + k * THREADS;            // float4 index, coalesced
        const int patch = idx / V4_PER_PATCH;           // 0..31
        // Flat LDS addresses keep the byte offset in the low 32 bits.
        uint32_t lds_addr = (uint32_t)(uintptr_t)(&sdataf[(idx + patch) * 4]);
        uint64_t gaddr    = (uint64_t)(uintptr_t)(gsrc + idx);
        asm volatile("global_load_async_to_lds_b128 %0, %1, off"
                     :: "v"(lds_addr), "v"(gaddr)
                     : "memory");
    }
    asm volatile("s_wait_asynccnt 0" ::: "memory");     // this wave's DMAs done
    __syncthreads();                                    // visible to all waves

    // ---------------- Phase 2: per-patch mean / unbiased std ----------------
    // 8 threads per patch; each handles 8 whole points (24 floats = 6 float4).
    {
        const int p = tid >> 3;   // patch 0..31
        const int s = tid & 7;    // subgroup lane 0..7
        float acc[3]  = {0.f, 0.f, 0.f};
        float accq[3] = {0.f, 0.f, 0.f};
        const float4* base =
            reinterpret_cast<const float4*>(&sdataf[(p * PATCH_STRIDE + s * 6) * 4]);
#pragma unroll
        for (int k = 0; k < 6; ++k) {
            float4 v = base[k];
            float f[4] = {v.x, v.y, v.z, v.w};
#pragma unroll
            for (int j = 0; j < 4; ++j) {
                const int c = (4 * k + j) % 3;          // constant after unroll
                acc[c]  += f[j];
                accq[c] += f[j] * f[j];
            }
        }
        // reduce across the 8-lane subgroup (wave32 shuffles)
#pragma unroll
        for (int m = 1; m < 8; m <<= 1) {
#pragma unroll
            for (int c = 0; c < 3; ++c) {
                acc[c]  += __shfl_xor(acc[c],  m, 32);
                accq[c] += __shfl_xor(accq[c], m, 32);
            }
        }
        if (s == 0) {
#pragma unroll
            for (int c = 0; c < 3; ++c) {
                const float mean = acc[c] * (1.0f / (float)PPP);
                float var = (accq[c] - acc[c] * acc[c] * (1.0f / (float)PPP))
                            * (1.0f / (float)(PPP - 1));
                var = fmaxf(var, 0.0f);                 // guard fp round-off
                smean[p][c] = mean;
                sstd[p][c]  = sqrtf(var);
            }
        }
    }
    __syncthreads();

    // ---------------- Phase 3: pairwise repulsion, upper triangle incl diag --
    float local = 0.f;
#pragma unroll
    for (int k = 0; k < (NPATCH * NPATCH) / THREADS; ++k) {
        const int q = tid + k * THREADS;   // 0..1023
        const int i = q >> 5;
        const int j = q & 31;
        if (i <= j) {
            float acc2 = 0.f;
#pragma unroll
            for (int c = 0; c < 3; ++c) {
                const float d  = fabsf(smean[i][c] - smean[j][c]);
                const float ss = sstd[i][c] + sstd[j][c];   // N_SIGMA == 1
                const float r  = fmaxf(ss - d, 0.f);
                acc2 += r * r;
            }
            local += sqrtf(acc2);
        }
    }

    // ---------------- block reduction -> partial (or atomic fallback) -------
#pragma unroll
    for (int m = 1; m < 32; m <<= 1) local += __shfl_xor(local, m, 32);
    if ((tid & 31) == 0) swsum[tid >> 5] = local;
    __syncthreads();
    if (tid == 0) {
        float t = 0.f;
#pragma unroll
        for (int w = 0; w < THREADS / 32; ++w) t += swsum[w];
        if (partials) {
            partials[b] = t;                  // deterministic two-stage path
        } else {
            atomicAdd(out, t * scale);        // fallback if d_ws too small
        }
    }
}

__global__ __launch_bounds__(THREADS)
void reduce_partials_kernel(const float* __restrict__ partials,
                            float* __restrict__ out,
                            int B, float scale) {
    __shared__ float swsum[THREADS / 32];
    float s = 0.f;
    for (int i = threadIdx.x; i < B; i += THREADS) s += partials[i];  // fixed order
#pragma unroll
    for (int m = 1; m < 32; m <<= 1) s += __shfl_xor(s, m, 32);
    if ((threadIdx.x & 31) == 0) swsum[threadIdx.x >> 5] = s;
    __syncthreads();
    if (threadIdx.x == 0) {
        float t = 0.f;
#pragma unroll
        for (int w = 0; w < THREADS / 32; ++w) t += swsum[w];
        out[0] = t * scale;
    }
}

extern "C" void kernel_launch(void* const* d_in, const int* in_sizes, int n_in,
                              void* d_out, int out_size, void* d_ws, size_t ws_size,
                              hipStream_t stream) {
    const float* pcs = (const float*)d_in[0];
    float* out = (float*)d_out;
    const int total = in_sizes[0];
    const int B = total / FLOATS_PER_B;
    if (B <= 0) return;

    const float n_pairs = (float)(NPATCH * (NPATCH + 1) / 2);  // 528
    const float scale = 1.0f / (n_pairs * (float)B);

    if (ws_size >= (size_t)B * sizeof(float)) {
        // Deterministic two-stage reduction through workspace.
        float* partials = (float*)d_ws;
        patch_repulsion_kernel<<<B, THREADS, 0, stream>>>(pcs, partials, out, scale);
        reduce_partials_kernel<<<1, THREADS, 0, stream>>>(partials, out, B, scale);
    } else {
        // Atomic fallback (still correct; fp-add order nondeterministic).
        hipMemsetAsync(out, 0, sizeof(float), stream);
        patch_repulsion_kernel<<<B, THREADS, 0, stream>>>(pcs, nullptr, out, scale);
    }
}